// S2Convolution_51084341019091
// MI455X (gfx1250) — compile-verified
//
#include <hip/hip_runtime.h>

typedef __attribute__((ext_vector_type(2))) float v2f;
typedef __attribute__((ext_vector_type(8))) float v8f;

#define PI_F 3.14159265358979323846f

// cumulative sum of (2l+1)^2  (Z-block offsets), and *32 (so3d offsets)
__constant__ int c_zoff[17] = {0,1,10,35,84,165,286,455,680,969,1330,1771,
                               2300,2925,3654,4495,5456};
__constant__ int c_soff[17] = {0,32,320,1120,2688,5280,9152,14560,21760,31008,
                               42560,56672,73600,93600,116928,143840,174592};

static __device__ inline v8f wmma4(v2f a, v2f b, v8f c) {
  // D = A(16x4 f32) * B(4x16 f32) + C ; fp32 WMMA, wave32
  return __builtin_amdgcn_wmma_f32_16x16x4_f32(false, a, false, b,
                                               (short)0, c, false, false);
}

// Wigner small-d element d^l_{mp,m}(beta), quantum norm, CS phase.
static __device__ float wig_d(int l, int mp, int m, float beta) {
  float c = cosf(0.5f * beta), s = sinf(0.5f * beta);
  float pref = 0.5f * (lgammaf((float)(l + mp + 1)) + lgammaf((float)(l - mp + 1)) +
                       lgammaf((float)(l + m + 1)) + lgammaf((float)(l - m + 1)));
  int lo = (m - mp) > 0 ? (m - mp) : 0;
  int hi = (l + m) < (l - mp) ? (l + m) : (l - mp);
  float v = 0.f;
  for (int k = lo; k <= hi; ++k) {
    float lg = pref - (lgammaf((float)(l + m - k + 1)) + lgammaf((float)(k + 1)) +
                       lgammaf((float)(mp - m + k + 1)) + lgammaf((float)(l - mp - k + 1)));
    float t = expf(lg) * powf(c, (float)(2 * l + m - mp - 2 * k))
                       * powf(s, (float)(mp - m + 2 * k));
    v += ((mp - m + k) & 1) ? -t : t;
  }
  return v;
}

// SOFT quadrature weight for b=32 (beta index j in 0..63), scaled by pi/32.
static __device__ float quad_w(int j) {
  float sum = 0.f;
  for (int k = 0; k < 32; ++k)
    sum += sinf((float)((2 * j + 1) * (2 * k + 1)) * (PI_F / 128.f)) / (float)(2 * k + 1);
  return (2.f / 32.f) * sinf((float)(2 * j + 1) * (PI_F / 128.f)) * sum * (PI_F / 32.f);
}

static __device__ inline int l_of_s(int s) {
  int l = 0;
  while ((l + 1) * (l + 1) <= s) ++l;
  return l;
}

// ---------------- constant-builder kernels ----------------

// wig[j*256+s] = w(beta_j) * d^l_{m,0}(beta_j),  beta_j = (j+0.5)/64*pi
__global__ void k_wig(float* __restrict__ wig) {
  int tid = blockIdx.x * blockDim.x + threadIdx.x;
  if (tid >= 64 * 256) return;
  int j = tid >> 8, s = tid & 255;
  int l = l_of_s(s);
  int m = s - l * l - l;
  float beta = ((float)j + 0.5f) * (PI_F / 64.f);
  wig[tid] = quad_w(j) * wig_d(l, m, 0, beta);
}

// W_t[s][j*64+a] = wig[j,s] * exp(-i*2*pi*m*a/64)   (s-major, K contiguous)
__global__ void k_W(const float* __restrict__ wig,
                    float* __restrict__ Wre, float* __restrict__ Wim) {
  int tid = blockIdx.x * blockDim.x + threadIdx.x;
  if (tid >= 256 * 4096) return;
  int s = tid >> 12, ja = tid & 4095;
  int j = ja >> 6, a = ja & 63;
  int l = l_of_s(s);
  int m = s - l * l - l;
  float wv = wig[j * 256 + s];
  float ang = (2.f * PI_F / 64.f) * (float)(m * a);
  Wre[tid] = wv * cosf(ang);
  Wim[tid] = -wv * sinf(ang);
}

// F[g*256+s] = sqrt((2l+1)/4pi) * exp(-i m alpha_g) * d^l_{m,0}(beta_g)
__global__ void k_F(float* __restrict__ Fre, float* __restrict__ Fim) {
  int tid = blockIdx.x * blockDim.x + threadIdx.x;
  if (tid >= 24 * 256) return;
  int g = tid >> 8, s = tid & 255;
  int bi = g >> 3, ai = g & 7;
  float beta = (float)(bi + 1) * (PI_F / 24.f);       // (1..3)*MAX_BETA/3, MAX_BETA=pi/8
  float alpha = (2.f * PI_F / 8.f) * (float)ai;
  int l = l_of_s(s);
  int m = s - l * l - l;
  float amp = sqrtf((float)(2 * l + 1) / (4.f * PI_F));
  float d = wig_d(l, m, 0, beta);
  float ma = (float)m * alpha;
  Fre[tid] = amp * cosf(ma) * d;
  Fim[tid] = -amp * sinf(ma) * d;
}

// yc[s][o][i] = conj( sum_a kernel[i,o,a] * F[a,s] )   (o-major, i contiguous)
__global__ void k_yhat(const float* __restrict__ kern,
                       const float* __restrict__ Fre, const float* __restrict__ Fim,
                       float* __restrict__ ycr, float* __restrict__ yci) {
  int tid = blockIdx.x * blockDim.x + threadIdx.x;
  if (tid >= 256 * 64 * 64) return;
  int s = tid >> 12, rest = tid & 4095;
  int o = rest >> 6, i = rest & 63;
  const float* kp = kern + (size_t)(i * 64 + o) * 24;
  float yr = 0.f, yi = 0.f;
  for (int a = 0; a < 24; ++a) {
    float kv = kp[a];
    yr += kv * Fre[a * 256 + s];
    yi += kv * Fim[a * 256 + s];
  }
  ycr[tid] = yr;
  yci[tid] = -yi;   // conjugate
}

// so3d[soff(l) + j*(2l+1)^2 + (mp+l)*(2l+1) + (m+l)] = (2l+1)*d^l_{mp,m}(beta_j)
__global__ void k_so3d(float* __restrict__ so3d) {
  int tid = blockIdx.x * blockDim.x + threadIdx.x;
  if (tid >= 174592) return;
  int l = 0;
  while (tid >= c_soff[l + 1]) ++l;
  int rem = tid - c_soff[l];
  int dl = 2 * l + 1, dl2 = dl * dl;
  int j = rem / dl2, idx = rem % dl2;
  int mp = idx / dl - l, m = idx % dl - l;
  float beta = ((float)j + 0.5f) * (PI_F / 32.f);
  so3d[tid] = (float)dl * wig_d(l, mp, m, beta);
}

// ---------------- WMMA GEMM stages ----------------

// xhat[s][row] = sum_K x[row][K] * W[K][s] ; row=(b*64+f), K=(j*64+a)
// grid: (64 mtiles, 16 ntiles), block: 32 (one wave)
__global__ __launch_bounds__(32) void k_xhat(const float* __restrict__ x,
                                             const float* __restrict__ Wre,
                                             const float* __restrict__ Wim,
                                             float* __restrict__ xre,
                                             float* __restrict__ xim) {
  int mt = blockIdx.x, nt = blockIdx.y;
  int lane = threadIdx.x, r = lane & 15, h = lane >> 4;
  const float* ap  = x   + (size_t)(mt * 16 + r) * 4096 + 2 * h;
  const float* bre = Wre + (size_t)(nt * 16 + r) * 4096 + 2 * h;
  const float* bim = Wim + (size_t)(nt * 16 + r) * 4096 + 2 * h;
  v8f cre = {}, cim = {};
  for (int k = 0; k < 4096; k += 4) {
    v2f a  = *(const v2f*)(ap  + k);
    v2f br = *(const v2f*)(bre + k);
    v2f bi = *(const v2f*)(bim + k);
    cre = wmma4(a, br, cre);
    cim = wmma4(a, bi, cim);
  }
  int col = nt * 16 + r;
  for (int v = 0; v < 8; ++v) {
    int row = mt * 16 + v + 8 * h;
    xre[(size_t)col * 1024 + row] = cre[v];
    xim[(size_t)col * 1024 + row] = cim[v];
  }
}

// Z^l[b,o,m,n] = sum_i xhat[s_m,b,i] * yc[s_n,o,i]    (complex)
// grid: (5456 (l,m,n) pairs, 4 o-tiles), block: 32
__global__ __launch_bounds__(32) void k_z(const float* __restrict__ xre,
                                          const float* __restrict__ xim,
                                          const float* __restrict__ ycr,
                                          const float* __restrict__ yci,
                                          float* __restrict__ zre,
                                          float* __restrict__ zim) {
  int p = blockIdx.x, ot = blockIdx.y;
  int l = 0;
  while (p >= c_zoff[l + 1]) ++l;
  int q = p - c_zoff[l], dl = 2 * l + 1;
  int m = q / dl, n = q % dl;
  int sm = l * l + m, sn = l * l + n;
  int lane = threadIdx.x, r = lane & 15, h = lane >> 4;
  const float* axr = xre + (size_t)sm * 1024 + r * 64 + 2 * h;
  const float* axi = xim + (size_t)sm * 1024 + r * 64 + 2 * h;
  const float* byr = ycr + (size_t)sn * 4096 + (ot * 16 + r) * 64 + 2 * h;
  const float* byi = yci + (size_t)sn * 4096 + (ot * 16 + r) * 64 + 2 * h;
  v8f cre = {}, cim = {};
  for (int k = 0; k < 64; k += 4) {
    v2f ar = *(const v2f*)(axr + k);
    v2f ai = *(const v2f*)(axi + k);
    v2f br = *(const v2f*)(byr + k);
    v2f bi = *(const v2f*)(byi + k);
    cre = wmma4(ar, br, cre);
    cre = wmma4(-ai, bi, cre);
    cim = wmma4(ar, bi, cim);
    cim = wmma4(ai, br, cim);
  }
  size_t zbase = (size_t)c_zoff[l] + (size_t)m * dl + n;
  for (int v = 0; v < 8; ++v) {
    int b = v + 8 * h, o = ot * 16 + r;
    size_t idx = (size_t)(b * 64 + o) * 5456 + zbase;
    zre[idx] = cre[v];
    zim[idx] = cim[v];
  }
}

// Per (b,o): build S_j (32x32 spectral accumulator) in LDS, then inverse 2-D DFT
// (x N^2) via two WMMA matmul stages with on-the-fly twiddle fragments.
__global__ __launch_bounds__(32) void k_final(const float* __restrict__ zre,
                                              const float* __restrict__ zim,
                                              const float* __restrict__ so3d,
                                              const float* __restrict__ bias,
                                              float* __restrict__ out) {
  __shared__ float zr_s[5456];
  __shared__ float zi_s[5456];
  __shared__ float Sre[32 * 32];
  __shared__ float Sim[32 * 32];
  __shared__ float Tre[32 * 32];
  __shared__ float Tim[32 * 32];

  int bo = blockIdx.x;
  int o = bo & 63;
  int lane = threadIdx.x, r = lane & 15, h = lane >> 4;
  const float tp32 = 2.f * PI_F / 32.f;

  for (int i = lane; i < 5456; i += 32) {
    zr_s[i] = zre[(size_t)bo * 5456 + i];
    zi_s[i] = zim[(size_t)bo * 5456 + i];
  }
  float bv = bias[o];
  __syncthreads();

  for (int j = 0; j < 32; ++j) {
    for (int i = lane; i < 1024; i += 32) { Sre[i] = 0.f; Sim[i] = 0.f; }
    __syncthreads();

    // S[mm mod 32][nn mod 32] = sum_l so3d^l_{mm,nn}(beta_j) * Z^l[mm,nn]
    for (int slot = lane; slot < 961; slot += 32) {
      int mm = slot / 31 - 15, nn = slot % 31 - 15;
      int am = mm < 0 ? -mm : mm, an = nn < 0 ? -nn : nn;
      int lmin = am > an ? am : an;
      float ar = 0.f, ai = 0.f;
      for (int l = lmin; l < 16; ++l) {
        int dl = 2 * l + 1;
        int idx = (mm + l) * dl + (nn + l);
        float d = so3d[c_soff[l] + j * dl * dl + idx];
        int zix = c_zoff[l] + idx;
        ar += d * zr_s[zix];
        ai += d * zi_s[zix];
      }
      Sre[(mm & 31) * 32 + (nn & 31)] = ar;
      Sim[(mm & 31) * 32 + (nn & 31)] = ai;
    }
    __syncthreads();

    // stage 1: T[m][g] = sum_n S[m][n] * exp(+i*2pi*n*g/32)
    for (int tm = 0; tm < 2; ++tm)
      for (int tg = 0; tg < 2; ++tg) {
        v8f tre = {}, tim = {};
        int grow = tg * 16 + r;   // gamma (B column)
        int srow = tm * 16 + r;   // m row  (A row)
        for (int k0 = 0; k0 < 32; k0 += 4) {
          v2f sre_f = *(const v2f*)&Sre[srow * 32 + k0 + 2 * h];
          v2f sim_f = *(const v2f*)&Sim[srow * 32 + k0 + 2 * h];
          float a0 = tp32 * (float)((k0 + 2 * h) * grow);
          float a1 = tp32 * (float)((k0 + 2 * h + 1) * grow);
          v2f cf = { __cosf(a0), __cosf(a1) };
          v2f sf = { __sinf(a0), __sinf(a1) };
          tre = wmma4(sre_f, cf, tre);
          tre = wmma4(-sim_f, sf, tre);
          tim = wmma4(sre_f, sf, tim);
          tim = wmma4(sim_f, cf, tim);
        }
        for (int v = 0; v < 8; ++v) {
          int mr = tm * 16 + v + 8 * h;
          int g = tg * 16 + r;
          Tre[mr * 32 + g] = tre[v];
          Tim[mr * 32 + g] = tim[v];
        }
      }
    __syncthreads();

    // stage 2: out[a][g] = Re( sum_m exp(+i*2pi*m*a/32) * T[m][g] )
    float* outp = out + ((size_t)bo * 32 + j) * 1024;
    for (int tm = 0; tm < 2; ++tm)
      for (int tg = 0; tg < 2; ++tg) {
        v8f acc = {};
        int arow = tm * 16 + r;   // alpha (A row)
        int g = tg * 16 + r;      // gamma (B column)
        for (int k0 = 0; k0 < 32; k0 += 4) {
          float a0 = tp32 * (float)((k0 + 2 * h) * arow);
          float a1 = tp32 * (float)((k0 + 2 * h + 1) * arow);
          v2f cf  = { __cosf(a0), __cosf(a1) };
          v2f nsf = { -__sinf(a0), -__sinf(a1) };
          v2f trf = { Tre[(k0 + 2 * h) * 32 + g], Tre[(k0 + 2 * h + 1) * 32 + g] };
          v2f tif = { Tim[(k0 + 2 * h) * 32 + g], Tim[(k0 + 2 * h + 1) * 32 + g] };
          acc = wmma4(cf, trf, acc);
          acc = wmma4(nsf, tif, acc);
        }
        for (int v = 0; v < 8; ++v) {
          int al = tm * 16 + v + 8 * h;
          outp[al * 32 + g] = acc[v] + bv;
        }
      }
    __syncthreads();
  }
}

extern "C" void kernel_launch(void* const* d_in, const int* in_sizes, int n_in,
                              void* d_out, int out_size, void* d_ws, size_t ws_size,
                              hipStream_t stream) {
  (void)in_sizes; (void)n_in; (void)out_size; (void)ws_size;
  const float* x    = (const float*)d_in[0];   // [16,64,64,64]
  const float* kern = (const float*)d_in[1];   // [64,64,24]
  const float* bias = (const float*)d_in[2];   // [1,64,1,1,1]
  float* out = (float*)d_out;                  // [16,64,32,32,32]
  float* ws = (float*)d_ws;

  size_t off = 0;
  float* Wre  = ws + off; off += 1048576;      // [256][4096]
  float* Wim  = ws + off; off += 1048576;
  float* wig  = ws + off; off += 16384;        // [64][256]
  float* Fre  = ws + off; off += 6144;         // [24][256]
  float* Fim  = ws + off; off += 6144;
  float* xre  = ws + off; off += 262144;       // [256][1024]
  float* xim  = ws + off; off += 262144;
  float* ycr  = ws + off; off += 1048576;      // [256][64][64]
  float* yci  = ws + off; off += 1048576;
  float* zr   = ws + off; off += 5586944;      // [1024][5456]
  float* zi   = ws + off; off += 5586944;
  float* so3d = ws + off; off += 174592;

  k_wig  <<<64,   256, 0, stream>>>(wig);
  k_W    <<<4096, 256, 0, stream>>>(wig, Wre, Wim);
  k_F    <<<24,   256, 0, stream>>>(Fre, Fim);
  k_yhat <<<4096, 256, 0, stream>>>(kern, Fre, Fim, ycr, yci);
  k_so3d <<<682,  256, 0, stream>>>(so3d);

  dim3 g1(64, 16);
  k_xhat<<<g1, 32, 0, stream>>>(x, Wre, Wim, xre, xim);
  dim3 g3(5456, 4);
  k_z<<<g3, 32, 0, stream>>>(xre, xim, ycr, yci, zr, zi);
  k_final<<<1024, 32, 0, stream>>>(zr, zi, so3d, bias, out);
}